// DenseControl_81793357185325
// MI455X (gfx1250) — compile-verified
//
#include <hip/hip_runtime.h>
#include <hip/hip_bf16.h>

typedef float v2f __attribute__((ext_vector_type(2)));
typedef float v4f __attribute__((ext_vector_type(4)));
typedef float v8f __attribute__((ext_vector_type(8)));

#define DIM 256
#define K_THRESH 100
#define NODES_PER_BLOCK 256

// ---------------------------------------------------------------------------
// Kernel 0: zero the accumulators in workspace (graph replays don't re-zero)
// ---------------------------------------------------------------------------
__global__ void init_acc_kernel(float* sumAcc, unsigned int* cnt) {
    int t = threadIdx.x;
    if (t < DIM) sumAcc[t] = 0.0f;
    if (t == DIM) *cnt = 0u;
}

// ---------------------------------------------------------------------------
// Kernel 1: masked column sum.  Each block scans NODES_PER_BLOCK rankings,
// builds a compact active list in LDS, then all 256 threads accumulate the
// active rows (one feature per thread).  Only blocks that saw an active node
// issue global atomics (~101 active rows total -> atomics are noise).
// ---------------------------------------------------------------------------
__global__ void masked_colsum_kernel(const float* __restrict__ x,
                                     const int* __restrict__ rank,
                                     float* __restrict__ sumAcc,
                                     unsigned int* __restrict__ cnt,
                                     int N) {
    __shared__ int activeList[NODES_PER_BLOCK];
    __shared__ int nActive;

    int tid = threadIdx.x;
    if (tid == 0) nActive = 0;
    __syncthreads();

    int node = blockIdx.x * NODES_PER_BLOCK + tid;
    if (node < N && rank[node] <= K_THRESH) {
        int slot = atomicAdd(&nActive, 1);
        activeList[slot] = node;
    }
    __syncthreads();

    int na = nActive;
    if (na == 0) return;

    float acc = 0.0f;
    for (int i = 0; i < na; ++i) {
        int n = activeList[i];
        acc += x[(size_t)n * DIM + tid];
    }
    atomicAdd(&sumAcc[tid], acc);
    if (tid == 0) atomicAdd(cnt, (unsigned int)na);
}

// ---------------------------------------------------------------------------
// Kernel 2: s = W @ m_x + count*b via V_WMMA_F32_16X16X4_F32 (fp32 WMMA).
// One wave (32 lanes).  B-tile broadcasts m_x across all 16 N-columns, so
// every column of the 16x16 accumulator equals the desired s segment.
//
// ISA layouts (CDNA5 ISA):
//   A 16x4 f32 : lane L -> M = L%16 ; VGPR0 holds K = (L<16 ? 0 : 2),
//                VGPR1 holds K = (L<16 ? 1 : 3)
//   B 4x16 f32 : lane L -> N = L%16 ; VGPR v holds K = v + 2*(L/16)
//   C/D 16x16  : lane L -> N = L%16 ; VGPR v holds M = v + 8*(L/16)
// ---------------------------------------------------------------------------
__global__ void gemv_wmma_kernel(const float* __restrict__ Wm,
                                 const float* __restrict__ bias,
                                 const float* __restrict__ mx,
                                 const unsigned int* __restrict__ cnt,
                                 float* __restrict__ s) {
    int lane = threadIdx.x;            // 0..31 (wave32, full wave)
    int m    = lane & 15;              // row within the 16-row tile
    int khi  = (lane >> 4) << 1;       // 0 for lanes 0-15, 2 for lanes 16-31
    float c  = (float)(*cnt);

    for (int mt = 0; mt < 16; ++mt) {
        v8f acc = {};
        const float* Wrow = Wm + (size_t)(mt * 16 + m) * DIM;
        for (int k = 0; k < DIM; k += 4) {
            v2f a, bb;
            a.x  = Wrow[k + khi];
            a.y  = Wrow[k + khi + 1];
            bb.x = mx[k + khi];
            bb.y = mx[k + khi + 1];
            // 8 args: (neg_a, A, neg_b, B, c_mod, C, reuse_a, reuse_b)
            acc = __builtin_amdgcn_wmma_f32_16x16x4_f32(
                false, a, false, bb, (short)0, acc, false, false);
        }
        // All 16 columns hold s; lanes with N==0 (lanes 0 and 16) each own
        // 8 distinct output rows: row = mt*16 + v + 8*(lane/16).
        if ((lane & 15) == 0) {
            int rbase = mt * 16 + ((lane >> 4) << 3);
            #pragma unroll
            for (int v = 0; v < 8; ++v) {
                int row = rbase + v;
                s[row] = acc[v] + c * bias[row];
            }
        }
    }
}

// ---------------------------------------------------------------------------
// Kernel 3: broadcast s to out[N, 256].  102.4 MB of stores -> this is the
// whole runtime.  128-bit nontemporal stores, fully coalesced: each group of
// 64 consecutive threads covers one row.
// ---------------------------------------------------------------------------
__global__ void broadcast_kernel(const float* __restrict__ s,
                                 float* __restrict__ out,
                                 int N) {
    int gtid = blockIdx.x * blockDim.x + threadIdx.x;
    int qcol = gtid & 63;                              // 128-bit column 0..63
    int row0 = gtid >> 6;
    int rowStride = (gridDim.x * blockDim.x) >> 6;

    const v4f* s4 = (const v4f*)s;
    v4f v = s4[qcol];
    v4f* out4 = (v4f*)out;

    for (int r = row0; r < N; r += rowStride) {
        __builtin_nontemporal_store(v, &out4[(size_t)r * 64 + qcol]);
    }
}

// ---------------------------------------------------------------------------
// Launch
// ---------------------------------------------------------------------------
extern "C" void kernel_launch(void* const* d_in, const int* in_sizes, int n_in,
                              void* d_out, int out_size, void* d_ws, size_t ws_size,
                              hipStream_t stream) {
    const float* x    = (const float*)d_in[0];
    // d_in[1] = edge_index : unused by the math (reference ignores it)
    const int*   rank = (const int*)d_in[2];
    const float* Wm   = (const float*)d_in[3];
    const float* bias = (const float*)d_in[4];
    float*       out  = (float*)d_out;

    int N = in_sizes[2];               // number of nodes

    // Workspace layout (floats): [0..255] m_x accumulator, [256] count (u32),
    // [320..575] s (byte offset 1280, 16B aligned).  Needs 2304 B of ws.
    float* ws            = (float*)d_ws;
    float* sumAcc        = ws;
    unsigned int* cnt    = (unsigned int*)(ws + DIM);
    float* s             = ws + 320;

    init_acc_kernel<<<1, 320, 0, stream>>>(sumAcc, cnt);

    int nBlocks = (N + NODES_PER_BLOCK - 1) / NODES_PER_BLOCK;
    masked_colsum_kernel<<<nBlocks, NODES_PER_BLOCK, 0, stream>>>(
        x, rank, sumAcc, cnt, N);

    gemv_wmma_kernel<<<1, 32, 0, stream>>>(Wm, bias, sumAcc, cnt, s);

    broadcast_kernel<<<1024, 256, 0, stream>>>(s, out, N);
}